// PairConvolution_1451698946405
// MI455X (gfx1250) — compile-verified
//
#include <hip/hip_runtime.h>

#define NPTS 64
#define NCI  32
#define NCO  32

typedef __attribute__((ext_vector_type(2))) float v2f;
typedef __attribute__((ext_vector_type(8))) float v8f;

// ---------------------------------------------------------------------------
// Kernel 1: per-z reductions + build tiny affine operator
//   ws layout per z (224 floats): A[3][32] @0, B[3][32] @96, C[32] @192
// out[z,a,b,i] = g[z,a,:]·A[z,:,i] + g[z,b,:]·B[z,:,i] + C[z,i]
// ---------------------------------------------------------------------------
__global__ void pairconv_reduce_kernel(const float* __restrict__ feat,
                                       const float* __restrict__ geom,
                                       const float* __restrict__ W,
                                       const int*   __restrict__ n_norm,
                                       float* __restrict__ ws) {
  const int z = blockIdx.x;
  const int t = threadIdx.x;

  __shared__ float fs[NPTS][NCI];   // sum over d
  __shared__ float fd[NPTS][NCI];   // sum over c
  __shared__ float ft[NCI];
  __shared__ float qs[3][NCI];
  __shared__ float qd[3][NCI];
  __shared__ float cpart[3][NCO];

  const float* fz = feat + (size_t)z * NPTS * NPTS * NCI;
  const float* gz = geom + (size_t)z * NPTS * 3;

  const int j  = t & (NCI - 1);
  const int rg = t >> 5;                         // 0..7

  // fs[c][j] = sum_d f[c,d,j]
  for (int c = rg; c < NPTS; c += 8) {
    float acc = 0.f;
    const float* p = fz + (size_t)c * NPTS * NCI + j;
    for (int d = 0; d < NPTS; ++d) acc += p[(size_t)d * NCI];
    fs[c][j] = acc;
  }
  // fd[d][j] = sum_c f[c,d,j]
  for (int d = rg; d < NPTS; d += 8) {
    float acc = 0.f;
    const float* p = fz + (size_t)d * NCI + j;
    for (int c = 0; c < NPTS; ++c) acc += p[(size_t)c * NPTS * NCI];
    fd[d][j] = acc;
  }
  __syncthreads();

  // ft, q_s, q_d  (one thread per channel j)
  if (t < NCI) {
    float tf = 0.f, s0 = 0.f, s1 = 0.f, s2 = 0.f, d0 = 0.f, d1 = 0.f, d2 = 0.f;
    for (int c = 0; c < NPTS; ++c) {
      const float vs = fs[c][t];
      const float vd = fd[c][t];
      const float gx = gz[c * 3 + 0], gy = gz[c * 3 + 1], gw = gz[c * 3 + 2];
      tf += vs;
      s0 += gx * vs; s1 += gy * vs; s2 += gw * vs;
      d0 += gx * vd; d1 += gy * vd; d2 += gw * vd;
    }
    ft[t] = tf;
    qs[0][t] = s0; qs[1][t] = s1; qs[2][t] = s2;
    qd[0][t] = d0; qd[1][t] = d1; qd[2][t] = d2;
  }
  __syncthreads();

  float* wsz = ws + (size_t)z * 224;
  const int   nn    = n_norm[0];
  const float scale = rsqrtf(6.0f * (float)(nn * nn));

  // One thread per (x,i): W[x, m*CO*CI + i*CI + j], m=0..5 -> k1..k6
  if (t < 3 * NCO) {
    const int x = t / NCO;
    const int i = t % NCO;
    const float* wb = W + (size_t)x * 6 * NCO * NCI + (size_t)i * NCI;
    float u0 = 0.f, u1 = 0.f, u2 = 0.f, u3 = 0.f, u4 = 0.f, cacc = 0.f;
    for (int jj = 0; jj < NCI; ++jj) {
      const float w0 = wb[0 * NCO * NCI + jj];
      const float w1 = wb[1 * NCO * NCI + jj];
      const float w2 = wb[2 * NCO * NCI + jj];
      const float w3 = wb[3 * NCO * NCI + jj];
      const float w4 = wb[4 * NCO * NCI + jj];
      const float w5 = wb[5 * NCO * NCI + jj];
      const float f  = ft[jj];
      u0 += w0 * f; u1 += w1 * f; u2 += w2 * f; u3 += w3 * f; u4 += w4 * f;
      cacc += (w1 + w3 - w5) * qs[x][jj] + (w2 + w4 + w5) * qd[x][jj];
    }
    wsz[x * NCO + i]      = -scale * (u0 + u1 + u2);   // A (coeff of g[a,x])
    wsz[96 + x * NCO + i] =  scale * (u0 - u3 - u4);   // B (coeff of g[b,x])
    cpart[x][i] = scale * cacc;
  }
  __syncthreads();
  if (t < NCO) wsz[192 + t] = cpart[0][t] + cpart[1][t] + cpart[2][t];  // C
}

// ---------------------------------------------------------------------------
// Kernel 2: out[(a,b), i] via V_WMMA_F32_16X16X4_F32.
// Per z: (4096 x 8) x (8 x 32) GEMM; row = [g_a0..2 | g_b0..2 | 1 | 0],
// operand rows = [A[0..2] | B[0..2] | C | 0]. One wave per 16-row tile.
// ---------------------------------------------------------------------------
__global__ void pairconv_wmma_kernel(const float* __restrict__ geom,
                                     const float* __restrict__ ws,
                                     float* __restrict__ out) {
  const int lane = threadIdx.x & 31;
  const int wave = threadIdx.x >> 5;
  const int tile = blockIdx.x * 8 + wave;   // 0..511
  const int z     = tile >> 8;              // /256 tiles per z
  const int trow  = tile & 255;
  const int a     = trow >> 2;              // 4 b-tiles per a
  const int btile = trow & 3;
  const int l16   = lane & 15;
  const int half  = lane >> 4;

  const float* gz  = geom + (size_t)z * NPTS * 3;
  const float* wsz = ws   + (size_t)z * 224;

  const int b = btile * 16 + l16;           // row M = l16 within tile
  const float ga0 = gz[a * 3 + 0], ga1 = gz[a * 3 + 1], ga2 = gz[a * 3 + 2];
  const float gb0 = gz[b * 3 + 0], gb1 = gz[b * 3 + 1], gb2 = gz[b * 3 + 2];

  // A-frag (16x4 f32): lane = M, half selects K pair {0,1}/{2,3} then {4,5}/{6,7}
  v2f a0, a1;
  a0.x = half ? ga2  : ga0;                 // K2 : K0
  a0.y = half ? gb0  : ga1;                 // K3 : K1
  a1.x = half ? 1.0f : gb1;                 // K6 : K4
  a1.y = half ? 0.0f : gb2;                 // K7 : K5

  // B-frags (4x16 f32): VGPR = low K row, half adds K/2; col N = i
  const int i0 = l16;                       // N-tile 0 -> i = 0..15
  const int i1 = 16 + l16;                  // N-tile 1 -> i = 16..31
  v2f b00, b01, b10, b11;
  b00.x = half ? wsz[64 + i0]  : wsz[i0];          // K2=A[2] : K0=A[0]
  b00.y = half ? wsz[96 + i0]  : wsz[32 + i0];     // K3=B[0] : K1=A[1]
  b01.x = half ? wsz[192 + i0] : wsz[128 + i0];    // K6=C    : K4=B[1]
  b01.y = half ? 0.0f          : wsz[160 + i0];    // K7=0    : K5=B[2]
  b10.x = half ? wsz[64 + i1]  : wsz[i1];
  b10.y = half ? wsz[96 + i1]  : wsz[32 + i1];
  b11.x = half ? wsz[192 + i1] : wsz[128 + i1];
  b11.y = half ? 0.0f          : wsz[160 + i1];

  v8f c0 = {};
  v8f c1 = {};
  c0 = __builtin_amdgcn_wmma_f32_16x16x4_f32(false, a0, false, b00, (short)0, c0, false, false);
  c0 = __builtin_amdgcn_wmma_f32_16x16x4_f32(false, a1, false, b01, (short)0, c0, false, false);
  c1 = __builtin_amdgcn_wmma_f32_16x16x4_f32(false, a0, false, b10, (short)0, c1, false, false);
  c1 = __builtin_amdgcn_wmma_f32_16x16x4_f32(false, a1, false, b11, (short)0, c1, false, false);

  // D-frag: VGPR v, lane -> M = v + 8*half, N = l16 (+0 / +16 per N-tile)
  float* ob = out + (((size_t)z * NPTS + a) * NPTS + (size_t)btile * 16) * NCO;
#pragma unroll
  for (int v = 0; v < 8; ++v) {
    const int M = v + 8 * half;
    ob[(size_t)M * NCO + l16]      = c0[v];
    ob[(size_t)M * NCO + 16 + l16] = c1[v];
  }
}

extern "C" void kernel_launch(void* const* d_in, const int* in_sizes, int n_in,
                              void* d_out, int out_size, void* d_ws, size_t ws_size,
                              hipStream_t stream) {
  const float* feat   = (const float*)d_in[0];   // [2,64,64,32]
  const float* geom   = (const float*)d_in[1];   // [2,64,3]
  const float* W      = (const float*)d_in[2];   // [3,6144]
  const int*   n_norm = (const int*)d_in[3];     // scalar
  float* out = (float*)d_out;                    // [2,64,64,32]
  float* ws  = (float*)d_ws;                     // needs 2*224 floats = 1792 B

  pairconv_reduce_kernel<<<2, 256, 0, stream>>>(feat, geom, W, n_norm, ws);
  pairconv_wmma_kernel<<<64, 256, 0, stream>>>(geom, ws, out);
}